// vPFAE_Graph_64690797412362
// MI455X (gfx1250) — compile-verified
//
#include <hip/hip_runtime.h>
#include <hip/hip_bf16.h>

// ---------------------------------------------------------------------------
// GNN VAE encoder for MI455X (gfx1250, wave32).
// Strategy: transform-then-aggregate (GEMMs via V_WMMA_F32_16X16X4_F32),
// edge scatter via global_atomic_add_f32, GCN head with rsqrt degree norm.
// ---------------------------------------------------------------------------

#define N_NODES 50000
#define IN_CH 128
#define L1_CH 96
#define L2_CH 64
#define OUT_CH 32
#define M_TILES (N_NODES / 16)   // 3125, exact

typedef __attribute__((ext_vector_type(2))) float v2f;
typedef __attribute__((ext_vector_type(8))) float v8f;

// ---------------------------------------------------------------------------
// WMMA GEMM: C[M x N] = A[M x K] @ B[K x N] (+ bias per column, optional).
// One wave computes a full 16-row strip (16 x N). B is staged in LDS in
// k-pair-interleaved layout: Bs[p*N + n] = (B[2p][n], B[2p+1][n]) so each
// B fragment is ONE aligned ds_load_b64 into an even VGPR pair (WMMA SRC1
// needs contiguous even-aligned registers; avoids v_mov shuffles).
//
// A-frag (32b 16x4): lanes 0-15 -> M=lane, v0=K0,v1=K1;
//                    lanes 16-31 -> M=lane-16, v0=K2,v1=K3.
// B-frag (32b 4x16): lanes 0-15 -> N=lane, v0=K0,v1=K1;
//                    lanes 16-31 -> N=lane-16, v0=K2,v1=K3.
// C/D: VGPR r, lanes 0-15 -> M=r, lanes 16-31 -> M=8+r; N=lane&15.
// ---------------------------------------------------------------------------
template <int K, int NT, bool BIAS>
__global__ __launch_bounds__(256) void gemm_wmma_f32(
    const float* __restrict__ A, const float* __restrict__ B,
    const float* __restrict__ bias, float* __restrict__ C) {
  constexpr int N = NT * 16;
  constexpr int KP = K / 2;           // number of k-pairs
  __shared__ v2f Bs[KP * N];          // k-pair interleaved B tile (<= 48 KB)

  // Cooperative fill: thread i handles (pair p, col n). Global reads are
  // coalesced (contiguous in n); LDS write is one b64 per thread.
  for (int i = threadIdx.x; i < KP * N; i += 256) {
    int p = i / N;
    int n = i - p * N;
    v2f pr;
    pr.x = B[(2 * p) * N + n];
    pr.y = B[(2 * p + 1) * N + n];
    Bs[p * N + n] = pr;
  }
  __syncthreads();

  const int waveInBlock = threadIdx.x >> 5;
  const int tileM = blockIdx.x * 8 + waveInBlock;
  if (tileM >= M_TILES) return;   // wave-uniform; EXEC stays all-ones for WMMA

  const int lane = threadIdx.x & 31;
  const int l = lane & 15;        // row (A) or column (B/C) within tile
  const int half = lane >> 4;     // selects K pair {0,1} vs {2,3}

  const float* Arow = A + (size_t)(tileM * 16 + l) * K;

  v8f acc[NT];
#pragma unroll
  for (int t = 0; t < NT; ++t) {
    float bv = BIAS ? bias[t * 16 + l] : 0.0f;
#pragma unroll
    for (int r = 0; r < 8; ++r) acc[t][r] = bv;
  }

#pragma unroll 4
  for (int kk = 0; kk < K; kk += 4) {
    // A fragment: single aligned global_load_b64 (kk and 2*half both even).
    v2f a = *(const v2f*)(Arow + kk + 2 * half);
    const v2f* brow = &Bs[(kk / 2 + half) * N];
#pragma unroll
    for (int t = 0; t < NT; ++t) {
      v2f b = brow[t * 16 + l];   // one ds_load_b64, even-aligned pair
      acc[t] = __builtin_amdgcn_wmma_f32_16x16x4_f32(
          /*neg_a=*/false, a, /*neg_b=*/false, b,
          /*c_mod=*/(short)0, acc[t], /*reuse_a=*/false, /*reuse_b=*/false);
    }
  }

#pragma unroll
  for (int t = 0; t < NT; ++t) {
#pragma unroll
    for (int r = 0; r < 8; ++r) {
      C[(size_t)(tileM * 16 + half * 8 + r) * N + t * 16 + l] = acc[t][r];
    }
  }
}

// ---------------------------------------------------------------------------
// Edge scatter-add: acc[dst[e]] += feat[src[e]], C channels, one wave/edge.
// ---------------------------------------------------------------------------
__global__ void scatter_edges(const float* __restrict__ feat,
                              float* __restrict__ acc,
                              const int* __restrict__ src,
                              const int* __restrict__ dst, int E, int C) {
  int gid = blockIdx.x * blockDim.x + threadIdx.x;
  int e = gid >> 5;
  int lane = gid & 31;
  if (e >= E) return;
  int s = src[e];
  int d = dst[e];
  const float* sp = feat + (size_t)s * C;
  float* dp = acc + (size_t)d * C;
  for (int c = lane; c < C; c += 32) unsafeAtomicAdd(&dp[c], sp[c]);
}

__global__ void relu_inplace(float* __restrict__ x, int n) {
  int i = blockIdx.x * blockDim.x + threadIdx.x;
  if (i < n) x[i] = fmaxf(x[i], 0.0f);
}

__global__ void deg_init(float* __restrict__ deg, int n) {
  int i = blockIdx.x * blockDim.x + threadIdx.x;
  if (i < n) deg[i] = 1.0f;  // self-loop
}

__global__ void deg_count(float* __restrict__ deg, const int* __restrict__ dst,
                          int E) {
  int e = blockIdx.x * blockDim.x + threadIdx.x;
  if (e < E) unsafeAtomicAdd(&deg[dst[e]], 1.0f);
}

__global__ void deg_rsqrt(float* __restrict__ deg, int n) {
  int i = blockIdx.x * blockDim.x + threadIdx.x;
  if (i < n) {
    float d = deg[i];
    deg[i] = (d > 0.0f) ? rsqrtf(d) : 0.0f;  // in place: now holds dinv
  }
}

// out[i][c] = b[c] + dinv[i]^2 * p[i][c]   (bias + self-loop message)
__global__ void gcn_init(const float* __restrict__ p_mu,
                         const float* __restrict__ p_ls,
                         const float* __restrict__ dinv,
                         const float* __restrict__ b_mu,
                         const float* __restrict__ b_ls,
                         float* __restrict__ out_mu,
                         float* __restrict__ out_ls, int n) {
  int gid = blockIdx.x * blockDim.x + threadIdx.x;
  if (gid >= n * OUT_CH) return;
  int i = gid >> 5;       // OUT_CH == 32
  int c = gid & 31;
  float s = dinv[i] * dinv[i];
  out_mu[gid] = b_mu[c] + s * p_mu[gid];
  out_ls[gid] = b_ls[c] + s * p_ls[gid];
}

__global__ void gcn_scatter(const float* __restrict__ p_mu,
                            const float* __restrict__ p_ls,
                            const float* __restrict__ dinv,
                            const int* __restrict__ src,
                            const int* __restrict__ dst,
                            float* __restrict__ out_mu,
                            float* __restrict__ out_ls, int E) {
  int gid = blockIdx.x * blockDim.x + threadIdx.x;
  int e = gid >> 5;
  int c = gid & 31;       // OUT_CH == 32 lanes exactly
  if (e >= E) return;
  int s = src[e];
  int d = dst[e];
  float coeff = dinv[s] * dinv[d];
  unsafeAtomicAdd(&out_mu[d * OUT_CH + c], coeff * p_mu[s * OUT_CH + c]);
  unsafeAtomicAdd(&out_ls[d * OUT_CH + c], coeff * p_ls[s * OUT_CH + c]);
}

// ---------------------------------------------------------------------------

extern "C" void kernel_launch(void* const* d_in, const int* in_sizes, int n_in,
                              void* d_out, int out_size, void* d_ws,
                              size_t ws_size, hipStream_t stream) {
  const float* x       = (const float*)d_in[0];
  const int*   ei      = (const int*)d_in[1];    // int32 edge_index [2, E]
  const float* w1_rel  = (const float*)d_in[2];
  const float* w1_root = (const float*)d_in[3];
  const float* b1      = (const float*)d_in[4];
  const float* w2_rel  = (const float*)d_in[5];
  const float* w2_root = (const float*)d_in[6];
  const float* b2      = (const float*)d_in[7];
  const float* w_mu    = (const float*)d_in[8];
  const float* b_mu    = (const float*)d_in[9];
  const float* w_ls    = (const float*)d_in[10];
  const float* b_ls    = (const float*)d_in[11];

  const int E = in_sizes[1] / 2;
  const int* src = ei;
  const int* dst = ei + E;

  float* out_mu = (float*)d_out;
  float* out_ls = out_mu + (size_t)N_NODES * OUT_CH;

  // Workspace layout (floats). Every region is fully written before read.
  float* t_buf = (float*)d_ws;                         // N*96 (reused N*64)
  float* h1    = t_buf + (size_t)N_NODES * L1_CH;      // N*96
  float* h2    = h1    + (size_t)N_NODES * L1_CH;      // N*64
  float* p_mu  = h2    + (size_t)N_NODES * L2_CH;      // N*32
  float* p_ls  = p_mu  + (size_t)N_NODES * OUT_CH;     // N*32
  float* dinv  = p_ls  + (size_t)N_NODES * OUT_CH;     // N   (deg then dinv)

  const dim3 blk(256);
  const int gemmBlocks   = (M_TILES + 7) / 8;                // 391
  const int edgeWaveBlks = (E * 32 + 255) / 256;             // 1 wave / edge
  const int edgeBlks     = (E + 255) / 256;
  const int nodeBlks     = (N_NODES + 255) / 256;

  // ---- Layer 1: GraphConv(128 -> 96) ----
  // t1 = x @ w1_rel ; h1 = x @ w1_root + b1 ; h1 += scatter(t1) ; relu
  gemm_wmma_f32<IN_CH, L1_CH / 16, false><<<gemmBlocks, blk, 0, stream>>>(
      x, w1_rel, nullptr, t_buf);
  gemm_wmma_f32<IN_CH, L1_CH / 16, true><<<gemmBlocks, blk, 0, stream>>>(
      x, w1_root, b1, h1);
  scatter_edges<<<edgeWaveBlks, blk, 0, stream>>>(t_buf, h1, src, dst, E, L1_CH);
  relu_inplace<<<(N_NODES * L1_CH + 255) / 256, blk, 0, stream>>>(
      h1, N_NODES * L1_CH);

  // ---- Layer 2: GraphConv(96 -> 64) ----
  gemm_wmma_f32<L1_CH, L2_CH / 16, false><<<gemmBlocks, blk, 0, stream>>>(
      h1, w2_rel, nullptr, t_buf);
  gemm_wmma_f32<L1_CH, L2_CH / 16, true><<<gemmBlocks, blk, 0, stream>>>(
      h1, w2_root, b2, h2);
  scatter_edges<<<edgeWaveBlks, blk, 0, stream>>>(t_buf, h2, src, dst, E, L2_CH);
  relu_inplace<<<(N_NODES * L2_CH + 255) / 256, blk, 0, stream>>>(
      h2, N_NODES * L2_CH);

  // ---- GCN normalization: deg = 1 + indeg ; dinv = rsqrt(deg) ----
  deg_init<<<nodeBlks, blk, 0, stream>>>(dinv, N_NODES);
  deg_count<<<edgeBlks, blk, 0, stream>>>(dinv, dst, E);
  deg_rsqrt<<<nodeBlks, blk, 0, stream>>>(dinv, N_NODES);

  // ---- GCN head: p = h2 @ w ; out = b + self-loop + edge scatter ----
  gemm_wmma_f32<L2_CH, OUT_CH / 16, false><<<gemmBlocks, blk, 0, stream>>>(
      h2, w_mu, nullptr, p_mu);
  gemm_wmma_f32<L2_CH, OUT_CH / 16, false><<<gemmBlocks, blk, 0, stream>>>(
      h2, w_ls, nullptr, p_ls);
  gcn_init<<<(N_NODES * OUT_CH + 255) / 256, blk, 0, stream>>>(
      p_mu, p_ls, dinv, b_mu, b_ls, out_mu, out_ls, N_NODES);
  gcn_scatter<<<edgeWaveBlks, blk, 0, stream>>>(p_mu, p_ls, dinv, src, dst,
                                                out_mu, out_ls, E);
}